// ClusterLoss_71116068487517
// MI455X (gfx1250) — compile-verified
//
#include <hip/hip_runtime.h>

typedef float v2f __attribute__((ext_vector_type(2)));
typedef float v8f __attribute__((ext_vector_type(8)));

namespace {
constexpr float kInvTemp = 1.0f / 0.07f;
constexpr float kAlpha   = 0.4f;
constexpr float kBeta    = 0.2f;
constexpr float kLambda  = 0.2f;
constexpr float kNegInf  = -1e30f;
constexpr int   kD       = 512;
constexpr int   kNMax    = 256;
constexpr int   kMMax    = 32;
constexpr int   kBlock   = 256;                // 8 wave32s
constexpr int   kGroups  = (kNMax / 16) + (kMMax / 16);  // 16 hn + 2 fn = 18
}

__global__ void __launch_bounds__(kBlock) cluster_loss_init(float* wsf, int* wsi) {
  wsf[0] = 0.0f; wsf[1] = 0.0f; wsf[2] = 0.0f;
  wsi[0] = 0;    wsi[1] = 0;
}

__global__ void __launch_bounds__(kBlock) cluster_loss_main(
    const float* __restrict__ q,  const float* __restrict__ k,
    const float* __restrict__ k2, const float* __restrict__ hn,
    const float* __restrict__ fnm, const int* __restrict__ hn_counts,
    const int* __restrict__ fn_counts,
    float* __restrict__ wsf, int* __restrict__ wsi)
{
  __shared__ float qs[kD];          // staged q row
  __shared__ float s_lneg[kNMax];   // raw hn dot products
  __shared__ float s_fnd[kMMax];    // raw fn dot products
  __shared__ float sA[kBlock];
  __shared__ float sB[kBlock];
  __shared__ float s_qk, s_qk2;

  const int b    = blockIdx.x;
  const int t    = threadIdx.x;
  const int lane = t & 31;
  const int wave = t >> 5;
  const int kh   = lane >> 4;   // lane half: 0 -> K0/K1, 1 -> K2/K3
  const int lr   = lane & 15;   // row within 16-row tile
  const int koff = kh * 2;

  // ---- stage q[b,:] into LDS (one 2KB row, reused ~288x) ----
  const size_t qbase = (size_t)b * kD;
  qs[t]          = q[qbase + t];
  qs[t + kBlock] = q[qbase + t + kBlock];
  __syncthreads();

  // ---- positive dots q.k and q.k2 (block reduction) ----
  {
    float p1 = qs[t] * k[qbase + t]  + qs[t + kBlock] * k[qbase + t + kBlock];
    float p2 = qs[t] * k2[qbase + t] + qs[t + kBlock] * k2[qbase + t + kBlock];
    sA[t] = p1; sB[t] = p2;
    __syncthreads();
    for (int s = kBlock / 2; s > 0; s >>= 1) {
      if (t < s) { sA[t] += sA[t + s]; sB[t] += sB[t + s]; }
      __syncthreads();
    }
    if (t == 0) { s_qk = sA[0]; s_qk2 = sB[0]; }
    __syncthreads();
  }

  // ---- streaming WMMA matvec: 16-row tiles of hn/fn vs broadcast q ----
  // A (16x4 f32): lane lr rows; lanes 0-15 carry K0,K1; lanes 16-31 carry K2,K3.
  // B (4x16 f32): broadcast q chunk, low-K in lanes 0-15, high-K in lanes 16-31.
  for (int g = wave; g < kGroups; g += (kBlock / 32)) {
    const float* rowp;
    float* dst;
    if (g < kNMax / 16) {
      rowp = hn  + ((size_t)b * kNMax + g * 16 + lr) * kD;
      dst  = s_lneg + g * 16;
    } else {
      rowp = fnm + ((size_t)b * kMMax + (g - kNMax / 16) * 16 + lr) * kD;
      dst  = s_fnd + (g - kNMax / 16) * 16;
    }
    v8f acc = {};
#pragma unroll 4
    for (int d = 0; d < kD; d += 4) {
      v2f av = *(const v2f*)(rowp + d + koff);   // global_load_b64, streaming
      v2f bv = *(const v2f*)(qs   + d + koff);   // ds_load_b64
      acc = __builtin_amdgcn_wmma_f32_16x16x4_f32(
          false, av, false, bv, (short)0, acc, false, false);
    }
    // C/D layout: VGPR r, lanes 0-15 -> M=r (N=lane), lanes 16-31 -> M=8+r.
    // Column N=0 lives in lanes 0 and 16.
    if (lr == 0) {
#pragma unroll
      for (int r = 0; r < 8; ++r) dst[kh * 8 + r] = acc[r];
    }
  }
  __syncthreads();

  // ---- masked logsumexp over the 256 negative logits ----
  const int cnt = hn_counts[b];
  const float v = (t < cnt) ? s_lneg[t] * kInvTemp : kNegInf;
  sA[t] = v;
  __syncthreads();
  for (int s = kBlock / 2; s > 0; s >>= 1) {
    if (t < s) sA[t] = fmaxf(sA[t], sA[t + s]);
    __syncthreads();
  }
  const float mx = sA[0];
  __syncthreads();
  sA[t] = (t < cnt) ? __expf(v - mx) : 0.0f;
  __syncthreads();
  for (int s = kBlock / 2; s > 0; s >>= 1) {
    if (t < s) sA[t] += sA[t + s];
    __syncthreads();
  }

  // ---- fused epilogue: CE terms + BML hinge, accumulate via atomics ----
  if (t == 0 && cnt > 0) {
    const float lse = mx + __logf(sA[0]);
    const float qk  = s_qk;
    const float qk2 = s_qk2;
    const float lp  = qk  * kInvTemp;
    const float lpn = qk2 * kInvTemp;
    const float m1  = fmaxf(lp, lse);
    const float ce  = m1 + __logf(__expf(lp  - m1) + __expf(lse - m1)) - lp;
    const float m2  = fmaxf(lpn, lse);
    const float cen = m2 + __logf(__expf(lpn - m2) + __expf(lse - m2)) - lpn;
    atomicAdd(&wsf[0], ce);
    atomicAdd(&wsf[1], cen);
    atomicAdd(&wsi[0], 1);

    const int cfn = fn_counts[b];
    if (cfn > 0) {
      float sfn = 0.0f;
      for (int m = 0; m < cfn; ++m) sfn += s_fnd[m];
      const float delta = sfn / (float)cfn - qk;
      const float bml = fmaxf(delta + kAlpha, 0.0f) + fmaxf(-delta - kBeta, 0.0f);
      atomicAdd(&wsf[2], bml);
      atomicAdd(&wsi[1], 1);
    }
  }
}

__global__ void __launch_bounds__(64) cluster_loss_final(
    const float* __restrict__ wsf, const int* __restrict__ wsi,
    float* __restrict__ out)
{
  if (threadIdx.x != 0 || blockIdx.x != 0) return;
  const int nvi = wsi[0] > 1 ? wsi[0] : 1;
  const float nv  = (float)nvi;
  const float cl  = wsf[0] / nv;
  const float cln = wsf[1] / nv;
  const float bml = (wsi[1] > 0) ? kLambda * (wsf[2] / (float)wsi[1]) : 0.0f;
  out[0] = cl + cln + bml;
  out[1] = cl;
  out[2] = bml;
  out[3] = cln;
}

extern "C" void kernel_launch(void* const* d_in, const int* in_sizes, int n_in,
                              void* d_out, int out_size, void* d_ws, size_t ws_size,
                              hipStream_t stream) {
  const float* q   = (const float*)d_in[0];
  const float* k   = (const float*)d_in[1];
  const float* k2  = (const float*)d_in[2];
  const float* hn  = (const float*)d_in[3];
  const float* fnm = (const float*)d_in[4];
  const int* hnc   = (const int*)d_in[5];
  const int* fnc   = (const int*)d_in[6];

  float* wsf = (float*)d_ws;
  int*   wsi = (int*)((char*)d_ws + 4 * sizeof(float));
  float* out = (float*)d_out;

  const int B = in_sizes[5];  // 2048 samples, one workgroup each

  cluster_loss_init<<<1, 1, 0, stream>>>(wsf, wsi);
  cluster_loss_main<<<B, kBlock, 0, stream>>>(q, k, k2, hn, fnm, hnc, fnc, wsf, wsi);
  cluster_loss_final<<<1, 64, 0, stream>>>(wsf, wsi, out);
}